// ImprovedGraphSAGE_44822278701841
// MI455X (gfx1250) — compile-verified
//
#include <hip/hip_runtime.h>
#include <hip/hip_bf16.h>

// GraphSAGE forward for MI455X (gfx1250, wave32).
// mean_agg(h) @ Wl.T  ==  segment_sum((h @ Wl.T)[src]) / deg  -> all GEMMs are
// dense N x K @ K x 128 on the f32 WMMA pipe (V_WMMA_F32_16X16X4_F32), and the
// irregular part is a uniform 128-wide scatter-add that lives in L2 (51 MB
// accumulator << 192 MB L2). LDS tile staging for the K=128 layers goes
// through the Tensor Data Mover (tensor_load_to_lds + s_wait_tensorcnt), with
// the TDM pad feature reproducing the bank-conflict-skewed LDS stride.

typedef float v2f __attribute__((ext_vector_type(2)));
typedef float v8f __attribute__((ext_vector_type(8)));
typedef unsigned int u32x4 __attribute__((ext_vector_type(4)));
typedef int i32x4 __attribute__((ext_vector_type(4)));
typedef int i32x8 __attribute__((ext_vector_type(8)));

#define HD 128   // hidden width (== in_sizes[3])
#define RPB 64   // GEMM rows per block (4 x 16 M-subtiles)

#if __has_builtin(__builtin_amdgcn_tensor_load_to_lds) && \
    __has_builtin(__builtin_amdgcn_s_wait_tensorcnt)
#define HAVE_TDM 1
#else
#define HAVE_TDM 0
#endif

// ---------------------------------------------------------------------------
// TDM: async 2D tile load Global -> LDS (CDNA5 ISA).
// width_elems dwords per row, rows_tile rows into LDS, rows beyond rows_tensor
// read as zero (OOB). pad_enable inserts 4 pad dwords every 128 dwords ->
// LDS row stride = 132 floats (the GEMM's skewed stride for K=128).
__device__ __forceinline__ void tdm_load_2d_pad128(
    unsigned lds_byte_addr, const void* gptr, unsigned width_elems,
    unsigned rows_tile, unsigned rows_tensor, unsigned stride_elems) {
#if HAVE_TDM
  const unsigned long long ga = (unsigned long long)(uintptr_t)gptr;
  u32x4 g0;
  g0[0] = 1u;                                   // count=1, user mode
  g0[1] = lds_byte_addr;                        // LDS dest (bytes)
  g0[2] = (unsigned)ga;                         // global_addr[31:0]
  g0[3] = ((unsigned)(ga >> 32) & 0x01FFFFFFu)  // global_addr[56:32]
          | (2u << 30);                         // type=2 ("image")
  i32x8 g1;
  g1[0] = (int)((2u << 16)      // data_size = 4 B
                | (1u << 20)    // pad_enable
                | (6u << 22)    // pad_interval: 128 dwords
                | (3u << 25));  // pad_amount:   4 dwords
  g1[1] = (int)((width_elems & 0xFFFFu) << 16);                  // dim0[15:0]
  g1[2] = (int)((width_elems >> 16) | ((rows_tensor & 0xFFFFu) << 16));
  g1[3] = (int)((rows_tensor >> 16) | (width_elems << 16));      // tile_dim0
  g1[4] = (int)rows_tile;       // tile_dim1 (tile_dim2 = 0 -> 2D)
  g1[5] = (int)stride_elems;    // tensor_dim0_stride[31:0]
  g1[6] = 0;
  g1[7] = 0;
  i32x4 z4 = {0, 0, 0, 0};
#if defined(__clang_major__) && __clang_major__ >= 23
  i32x8 z8 = {0, 0, 0, 0, 0, 0, 0, 0};
  __builtin_amdgcn_tensor_load_to_lds(g0, g1, z4, z4, z8, 0);
#else
  __builtin_amdgcn_tensor_load_to_lds(g0, g1, z4, z4, 0);
#endif
#else
  (void)lds_byte_addr; (void)gptr; (void)width_elems;
  (void)rows_tile; (void)rows_tensor; (void)stride_elems;
#endif
}

__device__ __forceinline__ unsigned lds_off(const void* p) {
  // generic LDS pointer: addr[31:0] is the LDS byte address (ISA §10.2)
  return (unsigned)(uintptr_t)p;
}

// ---------------------------------------------------------------------------
// zero fill
__global__ void zero_f32(float* __restrict__ p, size_t n) {
  size_t i = (size_t)blockIdx.x * blockDim.x + threadIdx.x;
  size_t stride = (size_t)gridDim.x * blockDim.x;
  for (; i < n; i += stride) p[i] = 0.0f;
}

// ---------------------------------------------------------------------------
// degree: deg[dst[e]] += 1
__global__ void degree_kernel(const long long* __restrict__ dst,
                              float* __restrict__ deg, int E) {
  int e = blockIdx.x * blockDim.x + threadIdx.x;
  if (e >= E) return;
  atomicAdd(&deg[(size_t)dst[e]], 1.0f);
}

// ---------------------------------------------------------------------------
// GEMM: out[N x 128] = A[N x K] @ W[128 x K]^T  via V_WMMA_F32_16X16X4_F32.
//
// Block = 256 threads (8 waves); block owns 64 A-rows, wave w owns output
// columns [16w, 16w+16) and iterates 4 M-subtiles. W and the 64 x K A-tile are
// staged in LDS with zero-padded K (Kpad = ceil4(K)) and skewed row stride
// S = Kpad + 4 (16 lanes -> 16 distinct banks; v2f reads stay 8B-aligned).
// K == 128: tiles staged by the Tensor Data Mover (pad feature produces the
//           same stride-132 layout; OOB rows of the tail A-tile read as 0).
// K != 128: manual zero-padded staging loop.
// Inner loop is branch-free: EXEC stays all-ones (WMMA requirement); the
// 4x-unrolled k-loop lets adjacent fragment reads merge to ds_load_2addr_b64
// and software-pipeline against the WMMAs.
__global__ void gemm_nt_wmma(const float* __restrict__ A,
                             const float* __restrict__ W,
                             float* __restrict__ out, int Nrows, int K) {
  const int Kpad = (K + 3) & ~3;
  const int S = Kpad + 4;  // LDS row stride in floats
  extern __shared__ float smem[];
  float* Wsh = smem;            // [HD][S]
  float* Ash = smem + HD * S;   // [RPB][S]

  const int tid = threadIdx.x;
  const int m0g = blockIdx.x * RPB;

#if HAVE_TDM
  const bool use_tdm = (K == HD);
#else
  const bool use_tdm = false;
#endif
  if (use_tdm) {
    if (tid == 0) {
      const int remRows = Nrows - m0g;
      tdm_load_2d_pad128(lds_off(Wsh), W, HD, HD, HD, HD);
      tdm_load_2d_pad128(lds_off(Ash), A + (size_t)m0g * HD, HD, RPB,
                         (unsigned)(remRows < RPB ? remRows : RPB), HD);
#if HAVE_TDM
      __builtin_amdgcn_s_wait_tensorcnt(0);
#endif
    }
  } else {
    // stage W[128 x K], zero-padding k in [K, S)
    for (int i = tid; i < HD * S; i += 256) {
      const int n = i / S;
      const int k = i - n * S;
      Wsh[i] = (k < K) ? W[(size_t)n * K + k] : 0.0f;
    }
    // stage A-tile [64 x K], zero-padding pad-k and rows beyond Nrows
    for (int i = tid; i < RPB * S; i += 256) {
      const int r = i / S;
      const int k = i - r * S;
      const int row = m0g + r;
      Ash[i] = (k < K && row < Nrows) ? A[(size_t)row * K + k] : 0.0f;
    }
  }
  __syncthreads();

  const int wave = tid >> 5;
  const int lane = tid & 31;
  const int half = lane >> 4;     // 0: {K0,K1}, 1: {K2,K3}
  const int l16  = lane & 15;
  const int n0   = wave * 16;

  const float* Bp = Wsh + (n0 + l16) * S + half * 2;
  const float* Ap = Ash + l16 * S + half * 2;

  v8f c[4];
#pragma unroll
  for (int mt = 0; mt < 4; ++mt) c[mt] = (v8f){0, 0, 0, 0, 0, 0, 0, 0};

#pragma unroll 4
  for (int k0 = 0; k0 < Kpad; k0 += 4) {
    const v2f b  = *(const v2f*)(Bp + k0);
    const v2f a0 = *(const v2f*)(Ap + k0);
    const v2f a1 = *(const v2f*)(Ap + 16 * S + k0);
    const v2f a2 = *(const v2f*)(Ap + 32 * S + k0);
    const v2f a3 = *(const v2f*)(Ap + 48 * S + k0);
    c[0] = __builtin_amdgcn_wmma_f32_16x16x4_f32(false, a0, false, b, (short)0,
                                                 c[0], false, false);
    c[1] = __builtin_amdgcn_wmma_f32_16x16x4_f32(false, a1, false, b, (short)0,
                                                 c[1], false, false);
    c[2] = __builtin_amdgcn_wmma_f32_16x16x4_f32(false, a2, false, b, (short)0,
                                                 c[2], false, false);
    c[3] = __builtin_amdgcn_wmma_f32_16x16x4_f32(false, a3, false, b, (short)0,
                                                 c[3], false, false);
  }

  const int mbase = m0g + half * 8;
  if (m0g + RPB <= Nrows) {  // fast path: full block in range
#pragma unroll
    for (int mt = 0; mt < 4; ++mt)
#pragma unroll
      for (int r = 0; r < 8; ++r)
        out[(size_t)(mbase + mt * 16 + r) * HD + n0 + l16] = c[mt][r];
  } else {  // tail block
#pragma unroll
    for (int mt = 0; mt < 4; ++mt)
#pragma unroll
      for (int r = 0; r < 8; ++r) {
        const int m = mbase + mt * 16 + r;
        if (m < Nrows) out[(size_t)m * HD + n0 + l16] = c[mt][r];
      }
  }
}

// ---------------------------------------------------------------------------
// scatter: agg[dst[e]][:] += msg[src[e]][:]   (wave per edge, float4 per lane;
// one coalesced 512B gather per wave, 4 L2 atomics per lane)
__global__ void scatter_add128(const long long* __restrict__ src,
                               const long long* __restrict__ dst,
                               const float* __restrict__ msg,
                               float* __restrict__ agg, int E) {
  const int wid = blockIdx.x * (blockDim.x >> 5) + (threadIdx.x >> 5);
  if (wid >= E) return;
  const int lane = threadIdx.x & 31;
  // speculative prefetch of the upcoming edge-index stream (global_prefetch_b8;
  // OOB speculative prefetches are silently dropped by hardware)
  if (lane == 0) {
    __builtin_prefetch(src + wid + 2048, 0, 1);
    __builtin_prefetch(dst + wid + 2048, 0, 1);
  }
  const size_t s = (size_t)src[wid];
  const size_t d = (size_t)dst[wid];
  const float4 v = *(const float4*)(msg + s * HD + lane * 4);
  float* p = agg + d * HD + lane * 4;
  atomicAdd(p + 0, v.x);
  atomicAdd(p + 1, v.y);
  atomicAdd(p + 2, v.z);
  atomicAdd(p + 3, v.w);
}

// ---------------------------------------------------------------------------
// layer 0 epilogue: out = relu(agg/max(deg,1) + bl + hWr)
__global__ void finalize_relu(const float* __restrict__ agg,
                              const float* __restrict__ deg,
                              const float* __restrict__ bl,
                              const float* __restrict__ hWr,
                              float* __restrict__ out, int N) {
  size_t i = (size_t)blockIdx.x * blockDim.x + threadIdx.x;
  if (i >= (size_t)N * HD) return;
  const int node = (int)(i >> 7);
  const int c = (int)(i & (HD - 1));
  const float dm = fmaxf(deg[node], 1.0f);
  const float v = agg[i] / dm + bl[c] + hWr[i];
  out[i] = fmaxf(v, 0.0f);
}

// ---------------------------------------------------------------------------
// layers 1..2 epilogue: t = agg/deg + bl + hWr + res ; out = relu(LN(t))
// wave per node row, 4 features per lane, wave32 shuffle reductions.
__global__ void finalize_ln_relu(const float* __restrict__ agg,
                                 const float* __restrict__ deg,
                                 const float* __restrict__ bl,
                                 const float* __restrict__ hWr,
                                 const float* __restrict__ res,
                                 float* __restrict__ out, int N) {
  const int wid = blockIdx.x * (blockDim.x >> 5) + (threadIdx.x >> 5);
  if (wid >= N) return;
  const int lane = threadIdx.x & 31;
  const size_t base = (size_t)wid * HD + lane * 4;
  const float dm = fmaxf(deg[wid], 1.0f);

  float t[4];
  float s = 0.0f, s2 = 0.0f;
#pragma unroll
  for (int j = 0; j < 4; ++j) {
    const int c = lane * 4 + j;
    t[j] = agg[base + j] / dm + bl[c] + hWr[base + j] + res[base + j];
    s += t[j];
    s2 += t[j] * t[j];
  }
#pragma unroll
  for (int off = 16; off > 0; off >>= 1) {
    s += __shfl_xor(s, off, 32);
    s2 += __shfl_xor(s2, off, 32);
  }
  const float mu = s * (1.0f / HD);
  const float var = s2 * (1.0f / HD) - mu * mu;
  const float inv = rsqrtf(var + 1e-5f);
#pragma unroll
  for (int j = 0; j < 4; ++j)
    out[base + j] = fmaxf((t[j] - mu) * inv, 0.0f);
}

// ---------------------------------------------------------------------------
// final linear: out[i][0..1] = h[i] . Wlin[0..1] + blin   (wave per node)
__global__ void final_linear(const float* __restrict__ h,
                             const float* __restrict__ Wlin,
                             const float* __restrict__ blin,
                             float* __restrict__ out, int N) {
  const int wid = blockIdx.x * (blockDim.x >> 5) + (threadIdx.x >> 5);
  if (wid >= N) return;
  const int lane = threadIdx.x & 31;
  const float4 v  = *(const float4*)(h + (size_t)wid * HD + lane * 4);
  const float4 w0 = *(const float4*)(Wlin + lane * 4);
  const float4 w1 = *(const float4*)(Wlin + HD + lane * 4);
  float d0 = v.x * w0.x + v.y * w0.y + v.z * w0.z + v.w * w0.w;
  float d1 = v.x * w1.x + v.y * w1.y + v.z * w1.z + v.w * w1.w;
#pragma unroll
  for (int off = 16; off > 0; off >>= 1) {
    d0 += __shfl_xor(d0, off, 32);
    d1 += __shfl_xor(d1, off, 32);
  }
  if (lane == 0) {
    out[(size_t)wid * 2 + 0] = d0 + blin[0];
    out[(size_t)wid * 2 + 1] = d1 + blin[1];
  }
}

// ---------------------------------------------------------------------------
extern "C" void kernel_launch(void* const* d_in, const int* in_sizes, int n_in,
                              void* d_out, int out_size, void* d_ws,
                              size_t ws_size, hipStream_t stream) {
  (void)n_in; (void)ws_size;
  const float* x        = (const float*)d_in[0];
  const long long* eidx = (const long long*)d_in[1];
  const float* Wl0 = (const float*)d_in[2];
  const float* bl0 = (const float*)d_in[3];
  const float* Wr0 = (const float*)d_in[4];
  const float* Wl1 = (const float*)d_in[5];
  const float* bl1 = (const float*)d_in[6];
  const float* Wr1 = (const float*)d_in[7];
  const float* Wl2 = (const float*)d_in[8];
  const float* bl2 = (const float*)d_in[9];
  const float* Wr2 = (const float*)d_in[10];
  const float* Wlin = (const float*)d_in[11];
  const float* blin = (const float*)d_in[12];

  const int Nn = out_size / 2;          // 100000
  const int F  = in_sizes[0] / Nn;      // 165
  const int E  = in_sizes[1] / 2;       // 1600000
  const long long* srcIdx = eidx;       // edge_index[0]
  const long long* dstIdx = eidx + E;   // edge_index[1]

  const size_t NH = (size_t)Nn * HD;
  const size_t nb = ((NH * sizeof(float)) + 255) & ~(size_t)255;
  char* ws = (char*)d_ws;
  float* bufA = (float*)(ws);            // h@Wl / next-h
  float* bufB = (float*)(ws + nb);       // h@Wr
  float* bufC = (float*)(ws + 2 * nb);   // agg (scatter accumulator)
  float* bufD = (float*)(ws + 3 * nb);   // current h
  float* deg  = (float*)(ws + 4 * nb);   // degree [N]

  const dim3 blk(256);
  const int mblocks   = (Nn + RPB - 1) / RPB;  // 1563 (last block 32 rows)
  const int eBlocks1  = (E + 255) / 256;       // thread-per-edge
  const int eBlocks8  = (E + 7) / 8;           // wave-per-edge (8 waves/blk)
  const int nhBlocks  = (int)((NH + 255) / 256);
  const int rowBlocks = (Nn + 7) / 8;          // wave-per-row (8 waves/blk)

  const int S0 = (((F + 3) & ~3) + 4);         // 172
  const int S1 = HD + 4;                       // 132
  const size_t shW0 = (size_t)(HD + RPB) * S0 * sizeof(float);  // 132,096 B
  const size_t shW  = (size_t)(HD + RPB) * S1 * sizeof(float);  // 101,376 B

  // degree (shared by all conv layers)
  zero_f32<<<(Nn + 255) / 256, blk, 0, stream>>>(deg, (size_t)Nn);
  degree_kernel<<<eBlocks1, blk, 0, stream>>>(dstIdx, deg, E);

  // ---- layer 0: h = relu(conv(x)) ----
  gemm_nt_wmma<<<mblocks, blk, shW0, stream>>>(x, Wl0, bufA, Nn, F);
  gemm_nt_wmma<<<mblocks, blk, shW0, stream>>>(x, Wr0, bufB, Nn, F);
  zero_f32<<<nhBlocks, blk, 0, stream>>>(bufC, NH);
  scatter_add128<<<eBlocks8, blk, 0, stream>>>(srcIdx, dstIdx, bufA, bufC, E);
  finalize_relu<<<nhBlocks, blk, 0, stream>>>(bufC, deg, bl0, bufB, bufD, Nn);

  // ---- layer 1: h = relu(LN(conv(h) + h)) ----  cur = bufD
  gemm_nt_wmma<<<mblocks, blk, shW, stream>>>(bufD, Wl1, bufA, Nn, HD);
  gemm_nt_wmma<<<mblocks, blk, shW, stream>>>(bufD, Wr1, bufB, Nn, HD);
  zero_f32<<<nhBlocks, blk, 0, stream>>>(bufC, NH);
  scatter_add128<<<eBlocks8, blk, 0, stream>>>(srcIdx, dstIdx, bufA, bufC, E);
  finalize_ln_relu<<<rowBlocks, blk, 0, stream>>>(bufC, deg, bl1, bufB, bufD,
                                                  bufA, Nn);

  // ---- layer 2 ----  cur = bufA
  gemm_nt_wmma<<<mblocks, blk, shW, stream>>>(bufA, Wl2, bufD, Nn, HD);
  gemm_nt_wmma<<<mblocks, blk, shW, stream>>>(bufA, Wr2, bufB, Nn, HD);
  zero_f32<<<nhBlocks, blk, 0, stream>>>(bufC, NH);
  scatter_add128<<<eBlocks8, blk, 0, stream>>>(srcIdx, dstIdx, bufD, bufC, E);
  finalize_ln_relu<<<rowBlocks, blk, 0, stream>>>(bufC, deg, bl2, bufB, bufA,
                                                  bufD, Nn);

  // ---- final linear ----  cur = bufD
  final_linear<<<rowBlocks, blk, 0, stream>>>(bufD, Wlin, blin, (float*)d_out,
                                              Nn);
}